// Net_44942537786163
// MI455X (gfx1250) — compile-verified
//
#include <hip/hip_runtime.h>

typedef __attribute__((ext_vector_type(2))) float v2f;
typedef __attribute__((ext_vector_type(8))) float v8f;

#define HEADS 8
#define NEG_SLOPE 0.2f

// ---------- ordered-uint mapping for float atomic max ----------
__device__ __forceinline__ unsigned f2ord(float f) {
    unsigned u = __float_as_uint(f);
    return (u & 0x80000000u) ? ~u : (u | 0x80000000u);
}
__device__ __forceinline__ float ord2f(unsigned k) {
    unsigned u = (k & 0x80000000u) ? (k & 0x7FFFFFFFu) : ~k;
    return __uint_as_float(u);
}
__device__ __forceinline__ float lrelu(float v) {
    return v > 0.f ? v : NEG_SLOPE * v;
}

// ---------- zero fill ----------
__global__ void k_zero(float* __restrict__ p, long n) {
    long i = (long)blockIdx.x * blockDim.x + threadIdx.x;
    long stride = (long)gridDim.x * blockDim.x;
    for (; i < n; i += stride) p[i] = 0.0f;
}

// ---------- GEMM via V_WMMA_F32_16X16X4_F32 ----------
// Out[nrows, NCOLS] = A[nrows, K] @ W[K, NCOLS], all row-major f32.
// Block = 128 threads (4 waves). Each block owns a 16-row tile (A staged in
// LDS); each wave computes 16x16 output tiles, looping column tiles by 4.
template <int K, int NCOLS>
__global__ __launch_bounds__(128) void k_gemm_wmma(const float* __restrict__ A,
                                                   const float* __restrict__ W,
                                                   float* __restrict__ Out,
                                                   int nrows) {
    __shared__ float sA[16 * K];
    const int row0 = blockIdx.x * 16;
    const int tid = threadIdx.x;

    // cooperative load of 16 x K A-tile (row-clamped) as float4
    {
        const float4* gA = reinterpret_cast<const float4*>(A);
        float4* lA = reinterpret_cast<float4*>(sA);
        const int kq = K / 4;
        for (int i = tid; i < 16 * kq; i += 128) {
            int r = i / kq, c = i % kq;
            int gr = row0 + r;
            if (gr >= nrows) gr = nrows - 1;
            lA[r * kq + c] = gA[(long)gr * kq + c];
        }
    }
    __syncthreads();

    const int lane = tid & 31;
    const int wave = tid >> 5;
    const int m = lane & 15;            // row within tile (A) / col within tile (B,C)
    const int kb = (lane >> 4) * 2;     // lanes 0-15: K 0,1 ; lanes 16-31: K 2,3

    for (int ct = wave; ct < NCOLS / 16; ct += 4) {
        const int col = ct * 16 + m;
        v8f acc = {};
#pragma unroll
        for (int k = 0; k < K; k += 4) {
            v2f a = *reinterpret_cast<const v2f*>(&sA[m * K + k + kb]);
            v2f b;
            b.x = W[(k + kb) * NCOLS + col];
            b.y = W[(k + kb + 1) * NCOLS + col];
            acc = __builtin_amdgcn_wmma_f32_16x16x4_f32(
                false, a, false, b, (short)0, acc, false, false);
        }
        // C/D layout: VGPR r -> lanes 0-15: M=r ; lanes 16-31: M=r+8 ; N=lane&15
        const int rbase = row0 + ((lane >> 4) << 3);
#pragma unroll
        for (int r = 0; r < 8; ++r) {
            int gr = rbase + r;
            if (gr < nrows) Out[(long)gr * NCOLS + col] = acc[r];
        }
    }
}

// ---------- per-node attention logits: a[n,h] = sum_c h[n,h,c]*att[h,c] ----------
__global__ void k_logits(const float* __restrict__ h, const float* __restrict__ att_s,
                         const float* __restrict__ att_d, float* __restrict__ as_,
                         float* __restrict__ ad_, int n, int C) {
    int idx = blockIdx.x * blockDim.x + threadIdx.x;
    if (idx >= n * HEADS) return;
    int node = idx / HEADS, hh = idx % HEADS;
    const float* hp = h + (long)node * HEADS * C + hh * C;
    float s = 0.f, d = 0.f;
    for (int c = 0; c < C; ++c) {
        float v = hp[c];
        s += v * att_s[hh * C + c];
        d += v * att_d[hh * C + c];
    }
    as_[idx] = s;
    ad_[idx] = d;
}

__device__ __forceinline__ void edge_decode(const int* __restrict__ ei, int E, int n,
                                            int e, int& s, int& d) {
    if (e < E) { s = ei[e]; d = ei[E + e]; }
    else       { s = e - E; d = e - E; }      // self loops appended
}

// ---------- pass 1: segment max of leaky-relu logits ----------
__global__ void k_edge_max(const int* __restrict__ ei, int E, int n,
                           const float* __restrict__ as_, const float* __restrict__ ad_,
                           unsigned* __restrict__ emax) {
    int e = blockIdx.x * blockDim.x + threadIdx.x;
    if (e >= E + n) return;
    int s, d;
    edge_decode(ei, E, n, e, s, d);
#pragma unroll
    for (int h = 0; h < HEADS; ++h) {
        float v = lrelu(as_[s * HEADS + h] + ad_[d * HEADS + h]);
        atomicMax(&emax[d * HEADS + h], f2ord(v));
    }
}

// ---------- pass 2: softmax denominator ----------
__global__ void k_edge_denom(const int* __restrict__ ei, int E, int n,
                             const float* __restrict__ as_, const float* __restrict__ ad_,
                             const unsigned* __restrict__ emax, float* __restrict__ den) {
    int e = blockIdx.x * blockDim.x + threadIdx.x;
    if (e >= E + n) return;
    int s, d;
    edge_decode(ei, E, n, e, s, d);
#pragma unroll
    for (int h = 0; h < HEADS; ++h) {
        float v = lrelu(as_[s * HEADS + h] + ad_[d * HEADS + h]);
        atomicAdd(&den[d * HEADS + h], __expf(v - ord2f(emax[d * HEADS + h])));
    }
}

// ---------- pass 3: weighted scatter-aggregate ----------
__global__ void k_edge_scatter(const int* __restrict__ ei, int E, int n,
                               const float* __restrict__ as_, const float* __restrict__ ad_,
                               const unsigned* __restrict__ emax, const float* __restrict__ den,
                               const float* __restrict__ h, float* __restrict__ agg, int C) {
    long idx = (long)blockIdx.x * blockDim.x + threadIdx.x;
    long total = (long)(E + n) * HEADS;
    if (idx >= total) return;
    int e = (int)(idx / HEADS);
    int hh = (int)(idx % HEADS);
    int s, d;
    edge_decode(ei, E, n, e, s, d);
    float v = lrelu(as_[s * HEADS + hh] + ad_[d * HEADS + hh]);
    float alpha = __expf(v - ord2f(emax[d * HEADS + hh])) / den[d * HEADS + hh];
    const float* hp = h + (long)s * HEADS * C + hh * C;
    float* op = agg + (long)d * HEADS * C + hh * C;
    for (int c = 0; c < C; ++c) atomicAdd(&op[c], alpha * hp[c]);
}

// ---------- bias + ELU (in place) ----------
__global__ void k_bias_elu(float* __restrict__ io, const float* __restrict__ b,
                           long total, int F) {
    long idx = (long)blockIdx.x * blockDim.x + threadIdx.x;
    if (idx >= total) return;
    float v = io[idx] + b[idx % F];
    io[idx] = v > 0.f ? v : expm1f(v);
}

// ---------- bias + log_softmax over 128 classes, one wave32 per row ----------
__global__ __launch_bounds__(256) void k_bias_logsoftmax(float* __restrict__ io,
                                                         const float* __restrict__ b,
                                                         int n) {
    int wid = (int)(((long)blockIdx.x * blockDim.x + threadIdx.x) >> 5);
    int lane = threadIdx.x & 31;
    if (wid >= n) return;
    float* row = io + (long)wid * 128;
    float v[4];
    float mx = -3.402823466e38f;
#pragma unroll
    for (int i = 0; i < 4; ++i) {
        v[i] = row[lane + 32 * i] + b[lane + 32 * i];
        mx = fmaxf(mx, v[i]);
    }
#pragma unroll
    for (int off = 16; off > 0; off >>= 1) mx = fmaxf(mx, __shfl_xor(mx, off, 32));
    float s = 0.f;
#pragma unroll
    for (int i = 0; i < 4; ++i) s += __expf(v[i] - mx);
#pragma unroll
    for (int off = 16; off > 0; off >>= 1) s += __shfl_xor(s, off, 32);
    float ls = __logf(s);
#pragma unroll
    for (int i = 0; i < 4; ++i) row[lane + 32 * i] = v[i] - mx - ls;
}

extern "C" void kernel_launch(void* const* d_in, const int* in_sizes, int n_in,
                              void* d_out, int out_size, void* d_ws, size_t ws_size,
                              hipStream_t stream) {
    const float* x   = (const float*)d_in[0];
    const int*   ei  = (const int*)d_in[1];
    const float* W1  = (const float*)d_in[2];
    const float* t1s = (const float*)d_in[3];
    const float* t1d = (const float*)d_in[4];
    const float* b1  = (const float*)d_in[5];
    const float* W2  = (const float*)d_in[6];
    const float* t2s = (const float*)d_in[7];
    const float* t2d = (const float*)d_in[8];
    const float* b2  = (const float*)d_in[9];

    const int n = in_sizes[0] / 128;
    const int E = in_sizes[1] / 2;

    // workspace layout (floats):  h (n*128, reused for both layers' projections)
    //                             as, ad, emax, den (n*8 each) | x2 (n*64)
    float* ws = (float*)d_ws;
    float*    h    = ws;                 ws += (long)n * 128;
    float*    as_  = ws;                 ws += (long)n * 8;
    float*    ad_  = ws;                 ws += (long)n * 8;
    unsigned* emax = (unsigned*)ws;      ws += (long)n * 8;
    float*    den  = ws;                 ws += (long)n * 8;
    float*    x2   = ws;                 ws += (long)n * 64;
    float*    out  = (float*)d_out;

    const int B = 256;
    const int rowTiles = (n + 15) / 16;
    const int ET = E + n;
    const int gE = (ET + B - 1) / B;
    const long scatTotal = (long)ET * HEADS;
    const int gScat = (int)((scatTotal + B - 1) / B);
    const int gNH = (n * HEADS + B - 1) / B;

    // ----- init accumulators (poisoned between calls by harness) -----
    k_zero<<<1024, B, 0, stream>>>((float*)emax, (long)n * 8);
    k_zero<<<1024, B, 0, stream>>>(den, (long)n * 8);
    k_zero<<<2048, B, 0, stream>>>(x2, (long)n * 64);
    k_zero<<<4096, B, 0, stream>>>(out, (long)n * 128);

    // ===== layer 1 =====
    k_gemm_wmma<128, 64><<<rowTiles, 128, 0, stream>>>(x, W1, h, n);
    k_logits<<<gNH, B, 0, stream>>>(h, t1s, t1d, as_, ad_, n, 8);
    k_edge_max<<<gE, B, 0, stream>>>(ei, E, n, as_, ad_, emax);
    k_edge_denom<<<gE, B, 0, stream>>>(ei, E, n, as_, ad_, emax, den);
    k_edge_scatter<<<gScat, B, 0, stream>>>(ei, E, n, as_, ad_, emax, den, h, x2, 8);
    k_bias_elu<<<(int)(((long)n * 64 + B - 1) / B), B, 0, stream>>>(x2, b1, (long)n * 64, 64);

    // ----- reset per-segment state for layer 2 -----
    k_zero<<<1024, B, 0, stream>>>((float*)emax, (long)n * 8);
    k_zero<<<1024, B, 0, stream>>>(den, (long)n * 8);

    // ===== layer 2 =====
    k_gemm_wmma<64, 128><<<rowTiles, 128, 0, stream>>>(x2, W2, h, n);
    k_logits<<<gNH, B, 0, stream>>>(h, t2s, t2d, as_, ad_, n, 16);
    k_edge_max<<<gE, B, 0, stream>>>(ei, E, n, as_, ad_, emax);
    k_edge_denom<<<gE, B, 0, stream>>>(ei, E, n, as_, ad_, emax, den);
    k_edge_scatter<<<gScat, B, 0, stream>>>(ei, E, n, as_, ad_, emax, den, h, out, 16);
    k_bias_logsoftmax<<<(n * 32 + B - 1) / B, B, 0, stream>>>(out, b2, n);
}